// MS_SSIM_23124103922416
// MI455X (gfx1250) — compile-verified
//
#include <hip/hip_runtime.h>
#include <hip/hip_bf16.h>

// ---------------------------------------------------------------------------
// MS-SSIM (5-level) on MI455X / gfx1250.
// Gaussian blur cast as banded GEMM via V_WMMA_F32_16X16X4_F32 (fp32 WMMA).
// One 16x16 output tile per wave32; fully fused blur + SSIM map + reduction.
// ---------------------------------------------------------------------------

typedef __attribute__((ext_vector_type(2))) float v2f;
typedef __attribute__((ext_vector_type(8))) float v8f;

#define WPB 4              // waves per block (128 threads)
#define WAVE_LDS_F 1904    // floats per wave: Xp(26*28) + Yp(26*28) + Hm(28*16)

__constant__ float GW[11] = {
    0.00102838f, 0.00759875f, 0.03600077f, 0.10936125f, 0.21300553f,
    0.26601263f,
    0.21300553f, 0.10936125f, 0.03600077f, 0.00759875f, 0.00102838f};

__device__ __forceinline__ float bandw(int j, int r) {
  // Banded Gaussian matrix entry: Band[j][r] = g[j-r] when 0 <= j-r <= 10.
  int d = j - r;
  return (d >= 0 && d < 11) ? GW[d] : 0.0f;
}

__device__ __forceinline__ float prodv(int arr, float x, float y) {
  switch (arr) {
    case 0: return x;
    case 1: return y;
    case 2: return x * x;
    case 3: return y * y;
    default: return x * y;
  }
}

__global__ void zero_accum_kernel(float* accum) {
  int t = threadIdx.x;
  if (t < 240) accum[t] = 0.0f;   // 4 levels of cs sums (4*48) + d sums (48)
}

// One wave computes one 16x16 valid-conv output tile for one (n,c) plane.
__global__ __launch_bounds__(128) void ssim_level_kernel(
    const float* __restrict__ X, const float* __restrict__ Y,
    int H, int W, float* accCS, float* accD, int last) {
  __shared__ float lds[WPB * WAVE_LDS_F];

  const int lane = threadIdx.x & 31;
  const int wave = threadIdx.x >> 5;
  const int OH = H - 10, OW = W - 10;
  const int tilesX = (OW + 15) >> 4;
  const int tilesY = (OH + 15) >> 4;
  const int nTiles = tilesX * tilesY;
  const int tile = blockIdx.x * WPB + wave;
  const int nc = blockIdx.y;
  if (tile >= nTiles) return;

  const int ty = tile / tilesX;
  const int tx = tile - ty * tilesX;
  const int r0 = ty << 4, c0 = tx << 4;

  float* Xp = lds + wave * WAVE_LDS_F;  // 26 rows x 28 cols (cols 26,27 = 0)
  float* Yp = Xp + 728;
  float* Hm = Yp + 728;                 // 28 rows x 16 cols (rows 26,27 = 0)

  const float* Xg = X + (size_t)nc * H * W;
  const float* Yg = Y + (size_t)nc * H * W;
  __builtin_prefetch(Xg + (size_t)r0 * W + c0, 0, 1);
  __builtin_prefetch(Yg + (size_t)r0 * W + c0, 0, 1);

  // Stage 26x26 clamped patch into LDS (padded to 28 wide with zeros).
  for (int i = lane; i < 26 * 28; i += 32) {
    int r = i / 28;
    int cc = i - r * 28;
    float xv = 0.0f, yv = 0.0f;
    if (cc < 26) {
      int gr = r0 + r; if (gr > H - 1) gr = H - 1;
      int gc = c0 + cc; if (gc > W - 1) gc = W - 1;
      xv = Xg[gr * W + gc];
      yv = Yg[gr * W + gc];
    }
    Xp[i] = xv;
    Yp[i] = yv;
  }
  Hm[416 + lane] = 0.0f;  // zero pad rows 26,27 of Hm (32 entries)

  const int mrow = lane & 15;            // M (A) / N (B,D) index for this lane
  const int khalf = (lane >> 4) << 1;    // K sub-offset: 0 (lanes 0-15) or 2

  // Band operand values are identical for the horizontal B-matrix and the
  // vertical A-matrix: precompute once per lane.
  v2f bw[7];
#pragma unroll
  for (int kk = 0; kk < 7; ++kk) {
    int j0 = kk * 4 + khalf;
    bw[kk].x = bandw(j0, mrow);
    bw[kk].y = bandw(j0 + 1, mrow);
  }

  v8f res[5];
#pragma unroll
  for (int arr = 0; arr < 5; ++arr) {
    // ---- horizontal pass: Hm(26x16) = P(26x26) x Band(26x16), 2 row tiles
#pragma unroll
    for (int t = 0; t < 2; ++t) {
      const int rb = t ? 10 : 0;
      v8f acc = {};
#pragma unroll
      for (int kk = 0; kk < 7; ++kk) {
        int idx = (rb + mrow) * 28 + kk * 4 + khalf;
        float x0 = Xp[idx], x1 = Xp[idx + 1];
        float y0 = Yp[idx], y1 = Yp[idx + 1];
        v2f a;
        a.x = prodv(arr, x0, y0);
        a.y = prodv(arr, x1, y1);
        acc = __builtin_amdgcn_wmma_f32_16x16x4_f32(
            false, a, false, bw[kk], (short)0, acc, false, false);
      }
      const int rowoff = rb + ((lane >> 4) << 3);
#pragma unroll
      for (int v = 0; v < 8; ++v) Hm[(rowoff + v) * 16 + mrow] = acc[v];
    }
    // ---- vertical pass: out(16x16) = BandV(16x26) x Hm(26x16)
    v8f acc = {};
#pragma unroll
    for (int kk = 0; kk < 7; ++kk) {
      int j0 = kk * 4 + khalf;
      v2f b;
      b.x = Hm[j0 * 16 + mrow];
      b.y = Hm[(j0 + 1) * 16 + mrow];
      acc = __builtin_amdgcn_wmma_f32_16x16x4_f32(
          false, bw[kk], false, b, (short)0, acc, false, false);
    }
    res[arr] = acc;
  }

  // ---- SSIM map + masked partial sums (8 pixels per lane)
  const float C1 = 0.0001f, C2 = 0.0009f;
  const int rbase = r0 + ((lane >> 4) << 3);
  float csSum = 0.0f, dSum = 0.0f;
#pragma unroll
  for (int v = 0; v < 8; ++v) {
    int orow = rbase + v, ocol = c0 + mrow;
    bool ok = (orow < OH) && (ocol < OW);
    float mu1 = res[0][v], mu2 = res[1][v];
    float s11 = res[2][v] - mu1 * mu1;
    float s22 = res[3][v] - mu2 * mu2;
    float s12 = res[4][v] - mu1 * mu2;
    float S2 = (2.0f * s12 + C2) / (s11 + s22 + C2);
    if (ok) {
      if (!last) {
        csSum += S2;
      } else {
        float S1 = (2.0f * mu1 * mu2 + C1) / (mu1 * mu1 + mu2 * mu2 + C1);
        float S = fminf(S1 + S2, 2.0f);
        dSum += sqrtf(fmaxf(2.0f - S, 0.0f));
      }
    }
  }
  // wave32 butterfly reduce
#pragma unroll
  for (int off = 16; off > 0; off >>= 1) {
    csSum += __shfl_xor(csSum, off, 32);
    dSum += __shfl_xor(dSum, off, 32);
  }
  if (lane == 0) {
    if (!last) atomicAdd(accCS + nc, csSum);
    else       atomicAdd(accD + nc, dSum);
  }
}

__global__ void avgpool2_kernel(const float* __restrict__ in,
                                float* __restrict__ out, int H, int W,
                                int total) {
  int i = blockIdx.x * 256 + threadIdx.x;
  if (i >= total) return;
  int Ho = H >> 1, Wo = W >> 1;
  int wo = i % Wo;
  int t = i / Wo;
  int ho = t % Ho;
  int p = t / Ho;
  const float* ip = in + (size_t)p * H * W + (size_t)(2 * ho) * W + 2 * wo;
  out[i] = 0.25f * (ip[0] + ip[1] + ip[W] + ip[W + 1]);
}

__global__ void finalize_kernel(const float* __restrict__ accum,
                                float* __restrict__ out) {
  __shared__ float sh[48];
  int t = threadIdx.x;
  if (t < 48) {
    const float w[5] = {0.0448f, 0.2856f, 0.3001f, 0.2363f, 0.1333f};
    const float npix[5] = {252004.0f, 60516.0f, 13924.0f, 2916.0f, 484.0f};
    float prod = 1.0f;
    for (int l = 0; l < 4; ++l) {
      float m = fmaxf(accum[l * 48 + t] / npix[l], 0.0f);
      prod *= powf(m, w[l]);
    }
    float md = fmaxf(accum[4 * 48 + t] / npix[4], 0.0f);
    prod *= powf(md, w[4]);
    sh[t] = prod;
  }
  __syncthreads();
  if (t == 0) {
    float s = 0.0f;
    for (int i = 0; i < 48; ++i) s += sh[i];
    out[0] = s / 48.0f;
  }
}

extern "C" void kernel_launch(void* const* d_in, const int* in_sizes, int n_in,
                              void* d_out, int out_size, void* d_ws,
                              size_t ws_size, hipStream_t stream) {
  const float* X = (const float*)d_in[0];
  const float* Y = (const float*)d_in[1];
  float* out = (float*)d_out;

  float* accum = (float*)d_ws;                 // 240 floats used
  float* XA = accum + 512;                     // level-1 sized: 16*3*256*256
  float* YA = XA + (size_t)16 * 3 * 256 * 256;
  float* XB = YA + (size_t)16 * 3 * 256 * 256; // level-2 sized: 16*3*128*128
  float* YB = XB + (size_t)16 * 3 * 128 * 128;

  zero_accum_kernel<<<1, 256, 0, stream>>>(accum);

  const int N = 16, C = 3;
  const float* curX = X;
  const float* curY = Y;
  int H = 512;
  for (int l = 0; l < 5; ++l) {
    int OH = H - 10;
    int tiles1d = (OH + 15) >> 4;
    int nT = tiles1d * tiles1d;
    dim3 grid((nT + WPB - 1) / WPB, N * C);
    ssim_level_kernel<<<grid, dim3(WPB * 32), 0, stream>>>(
        curX, curY, H, H, accum + l * 48, accum + 4 * 48, (l == 4) ? 1 : 0);
    if (l < 4) {
      float *nX, *nY;
      if (l == 0)      { nX = XA; nY = YA; }
      else if (l == 1) { nX = XB; nY = YB; }
      else if (l == 2) { nX = XA; nY = YA; }
      else             { nX = XB; nY = YB; }
      int Ho = H >> 1;
      int total = N * C * Ho * Ho;
      avgpool2_kernel<<<(total + 255) / 256, 256, 0, stream>>>(curX, nX, H, H,
                                                               total);
      avgpool2_kernel<<<(total + 255) / 256, 256, 0, stream>>>(curY, nY, H, H,
                                                               total);
      curX = nX;
      curY = nY;
      H = Ho;
    }
  }
  finalize_kernel<<<1, 64, 0, stream>>>(accum, out);
}